// SpatialTransformerPooled3d_3934190043443
// MI455X (gfx1250) — compile-verified
//
#include <hip/hip_runtime.h>
#include <hip/hip_bf16.h>
#include <stdint.h>

// SpatialTransformerPooled3d for MI455X (gfx1250, wave32).
//
// Roofline: dominant cost is streaming x once (128 MiB -> ~5.5us @ 23.3TB/s).
// Grid: 128 workgroups (one per b = n*16+t slab), 1024 threads (32 waves).
// Per channel: async-copy 16KB plane global->LDS (double buffered, ASYNCcnt),
// build avg-pool pyramid in LDS (L2->L3 via v_wmma_f32_16x16x4_f32 as
// L3 = P * L2 * P^T), then each thread bilinearly samples its <=2 points at
// all 4 levels and accumulates feat-weighted sums in registers.

#define AS1 __attribute__((address_space(1)))
#define AS3 __attribute__((address_space(3)))

typedef float v2f __attribute__((ext_vector_type(2)));
typedef float v8f __attribute__((ext_vector_type(8)));
typedef int   v4i __attribute__((ext_vector_type(4)));

#if __has_builtin(__builtin_amdgcn_wmma_f32_16x16x4_f32)
#define HAVE_WMMA_F32X4 1
#endif

__device__ __forceinline__ uint32_t lds_off_u32(const void* p) {
  return (uint32_t)(uintptr_t)(AS3 const void*)p;
}

// Each of the 1024 threads copies 16 bytes: one 64x64 f32 plane total.
__device__ __forceinline__ void async_copy_plane(const float* gsrc, float* ldst, int tid) {
#if __has_builtin(__builtin_amdgcn_global_load_async_to_lds_b128)
  __builtin_amdgcn_global_load_async_to_lds_b128(
      (AS1 v4i*)(gsrc + tid * 4), (AS3 v4i*)(ldst + tid * 4), 0, 0);
#else
  uint32_t l   = lds_off_u32(ldst + tid * 4);
  uint32_t off = (uint32_t)(tid * 16);
  uint64_t base = (uint64_t)(uintptr_t)gsrc;
  asm volatile("global_load_async_to_lds_b128 %0, %1, %2 offset:0"
               :: "v"(l), "v"(off), "s"(base) : "memory");
#endif
}

// 16x16 -> 8x8 average pool as L3 = P(8x16) * L2 * P^T, P[i,2i]=P[i,2i+1]=0.5,
// computed with v_wmma_f32_16x16x4_f32 (padded to 16x16) on wave 0.
__device__ __forceinline__ void l2_to_l3(const float* l2, float* l3, float* tls, int tid) {
#ifdef HAVE_WMMA_F32X4
  if (tid < 32) {
    const int lane = tid;
    const int m  = lane & 15;   // A-row / B-col / C-col index
    const int hi = lane >> 4;   // K sub-slot selector within fragments
    // Stage 1: T = P * L2   (M=16 rows, rows 8..15 of P are zero)
    v8f c = {};
#pragma unroll
    for (int kc = 0; kc < 4; ++kc) {
      const int k0 = kc * 4 + hi * 2;
      v2f a, b;
      a.x = (m < 8 && ((k0 >> 1) == m)) ? 0.5f : 0.0f;        // A[m,k0]
      a.y = (m < 8 && (((k0 + 1) >> 1) == m)) ? 0.5f : 0.0f;  // A[m,k0+1]
      b.x = l2[k0 * 16 + m];                                  // B[k0,  n=m]
      b.y = l2[(k0 + 1) * 16 + m];                            // B[k0+1,n=m]
      c = __builtin_amdgcn_wmma_f32_16x16x4_f32(false, a, false, b,
                                                (short)0, c, false, false);
    }
    // Spill T (16x16) to LDS to re-load as A fragments of stage 2.
#pragma unroll
    for (int v = 0; v < 8; ++v) tls[(v + hi * 8) * 16 + m] = c[v];
    // Stage 2: L3 = T * P^T ; P^T[k,n] = 0.5 iff n == k>>1
    v8f c2 = {};
#pragma unroll
    for (int kc = 0; kc < 4; ++kc) {
      const int k0 = kc * 4 + hi * 2;
      v2f a, b;
      a.x = tls[m * 16 + k0];
      a.y = tls[m * 16 + k0 + 1];
      b.x = (m == (k0 >> 1)) ? 0.5f : 0.0f;
      b.y = (m == ((k0 + 1) >> 1)) ? 0.5f : 0.0f;
      c2 = __builtin_amdgcn_wmma_f32_16x16x4_f32(false, a, false, b,
                                                 (short)0, c2, false, false);
    }
    if (lane < 8) {
#pragma unroll
      for (int v = 0; v < 8; ++v) l3[v * 8 + lane] = c2[v];  // L3[v][lane]
    }
  }
#else
  if (tid < 64) {
    const int oy = tid >> 3, ox = tid & 7;
    const float* r0 = l2 + (2 * oy) * 16 + 2 * ox;
    l3[tid] = 0.25f * (r0[0] + r0[1] + r0[16] + r0[17]);
  }
#endif
}

__global__ __launch_bounds__(1024, 1)
void stp3d_kernel(const float* __restrict__ x, const float* __restrict__ grid,
                  const float* __restrict__ feats, const float* __restrict__ bias,
                  float* __restrict__ out) {
  __shared__ float plane[2][4096];   // double-buffered 64x64 plane (32 KB)
  __shared__ float pyr[1344];        // L1 32x32 | L2 16x16 @1024 | L3 8x8 @1280
  __shared__ float tls[256];         // WMMA intermediate T (16x16)

  const int tid = (int)threadIdx.x;
  const int b   = (int)blockIdx.x;   // b = n*16 + t
  const int n   = b >> 4;
  const int tt  = b & 15;

  // ---- Precompute per-point, per-level bilinear params (register resident) ----
  float w00[2][4], w01[2][4], w10[2][4], w11[2][4];
  int   rel[2][4], dxo[2][4], dyo[2][4];
  float acc[2] = {0.0f, 0.0f};
  const int lvlBase[4] = {0, 0, 1024, 1280};  // float offsets within region
#pragma unroll
  for (int j = 0; j < 2; ++j) {
    const int p = tid + j * 1024;
    if (p < 2000) {
      float gx = fminf(fmaxf(grid[2 * p + 0], -1.0f), 1.0f);
      float gy = fminf(fmaxf(grid[2 * p + 1], -1.0f), 1.0f);
#pragma unroll
      for (int l = 0; l < 4; ++l) {
        const int Wl = 64 >> l;
        const float a = 0.5f * (float)(Wl - 1);
        const float fx = (gx + 1.0f) * a;
        const float fy = (gy + 1.0f) * a;
        const float fx0 = floorf(fx), fy0 = floorf(fy);
        const float wx = fx - fx0, wy = fy - fy0;
        int x0 = (int)fx0; x0 = max(0, min(x0, Wl - 1));
        const int x1 = min(x0 + 1, Wl - 1);
        int y0 = (int)fy0; y0 = max(0, min(y0, Wl - 1));
        const int y1 = min(y0 + 1, Wl - 1);
        rel[j][l] = (lvlBase[l] + y0 * Wl + x0) * 4;   // byte offset in region
        dxo[j][l] = (x1 - x0) * 4;
        dyo[j][l] = (y1 - y0) * Wl * 4;
        w00[j][l] = (1.0f - wy) * (1.0f - wx);
        w01[j][l] = (1.0f - wy) * wx;
        w10[j][l] = wy * (1.0f - wx);
        w11[j][l] = wy * wx;
      }
    }
  }

  // x[n, ch, tt, :, :] plane base; channel stride = 16*4096 elements
  const float* xbase = x + ((size_t)n * 1024 + (size_t)tt) * 4096;

  // Preload channel 0
  async_copy_plane(xbase, plane[0], tid);
  asm volatile("s_wait_asynccnt 0" ::: "memory");
  __syncthreads();

  for (int ch = 0; ch < 64; ++ch) {
    const int buf = ch & 1;
    if (ch + 1 < 64)
      async_copy_plane(xbase + (size_t)(ch + 1) * (16 * 4096), plane[buf ^ 1], tid);

    const float* pl = plane[buf];

    // L0(64x64) -> L1(32x32): 1024 threads, one output each
    {
      const int oy = tid >> 5, ox = tid & 31;
      const float* r = pl + oy * 128 + 2 * ox;
      pyr[tid] = 0.25f * (r[0] + r[1] + r[64] + r[65]);
    }
    __syncthreads();

    // L1(32x32) -> L2(16x16)
    if (tid < 256) {
      const int oy = tid >> 4, ox = tid & 15;
      const float* r = pyr + oy * 64 + 2 * ox;
      pyr[1024 + tid] = 0.25f * (r[0] + r[1] + r[32] + r[33]);
    }
    __syncthreads();

    // L2(16x16) -> L3(8x8) via WMMA (wave 0)
    l2_to_l3(pyr + 1024, pyr + 1280, tls, tid);
    __syncthreads();

    // Sample all 4 levels at this channel, accumulate feat-weighted sums
    const char* base0 = (const char*)pl;
    const char* base1 = (const char*)pyr;
#pragma unroll
    for (int j = 0; j < 2; ++j) {
      const int p = tid + j * 1024;
      if (p < 2000) {
#pragma unroll
        for (int l = 0; l < 4; ++l) {
          const char* a = ((l == 0) ? base0 : base1) + rel[j][l];
          const float v00 = *(const float*)(a);
          const float v01 = *(const float*)(a + dxo[j][l]);
          const float v10 = *(const float*)(a + dyo[j][l]);
          const float v11 = *(const float*)(a + dxo[j][l] + dyo[j][l]);
          const float s = v00 * w00[j][l] + v01 * w01[j][l] +
                          v10 * w10[j][l] + v11 * w11[j][l];
          acc[j] = fmaf(s, feats[(size_t)(l * 64 + ch) * 2000 + p], acc[j]);
        }
      }
    }

    // Next plane must be resident and pyramid consumed before overwrite.
    asm volatile("s_wait_asynccnt 0" ::: "memory");
    __syncthreads();
  }

#pragma unroll
  for (int j = 0; j < 2; ++j) {
    const int p = tid + j * 1024;
    if (p < 2000) out[(size_t)b * 2000 + p] = acc[j] + bias[p];
  }
}

extern "C" void kernel_launch(void* const* d_in, const int* in_sizes, int n_in,
                              void* d_out, int out_size, void* d_ws, size_t ws_size,
                              hipStream_t stream) {
  (void)in_sizes; (void)n_in; (void)d_ws; (void)ws_size; (void)out_size;
  const float* x     = (const float*)d_in[0];
  const float* grid  = (const float*)d_in[1];
  const float* feats = (const float*)d_in[2];
  const float* bias  = (const float*)d_in[3];
  float* out = (float*)d_out;
  hipLaunchKernelGGL(stp3d_kernel, dim3(128), dim3(1024), 0, stream,
                     x, grid, feats, bias, out);
}